// SumLayer_33234456937228
// MI455X (gfx1250) — compile-verified
//
#include <hip/hip_runtime.h>
#include <cstdint>
#include <cstddef>

// ---------------------------------------------------------------------------
// SumLayer weighted log-sum-exp (probabilistic-circuit sum nodes), gfx1250.
// G=8192 groups, C=64 children/group, B=128 batch. Memory bound (~264MB/call).
// CDNA5 paths used: global_load_async_to_lds_b128 (ASYNCcnt) for the 512B row
// gather, s_wait_asynccnt, and v_wmma_f32_16x16x4_f32 for the weighted
// child reduction.
// ---------------------------------------------------------------------------

typedef __attribute__((ext_vector_type(2))) float v2f;
typedef __attribute__((ext_vector_type(8))) float v8f;

static constexpr int C       = 64;    // children per group
static constexpr int B       = 128;   // batch size
static constexpr int RSTRIDE = 136;   // LDS row stride in floats (16B aligned,
                                      // 2-row delta = 16 banks -> conflict-free)

__device__ __forceinline__ void async_row_load_b128(uint32_t lds_byte_off,
                                                    const float* src) {
  // VDST = per-lane LDS byte address, ADDR = 64-bit global address, no saddr.
  asm volatile("global_load_async_to_lds_b128 %0, %1, off"
               :
               : "v"(lds_byte_off), "v"(src)
               : "memory");
}

__device__ __forceinline__ void wait_async_zero() {
#if __has_builtin(__builtin_amdgcn_s_wait_asynccnt)
  __builtin_amdgcn_s_wait_asynccnt(0);
#else
  asm volatile("s_wait_asynccnt 0" ::: "memory");
#endif
}

__global__ __launch_bounds__(256) void sum_layer_lse_kernel(
    const float* __restrict__ element_mars,  // [MAX_ELS, B]
    const float* __restrict__ params,        // [N_PARAMS]
    const int*   __restrict__ nids,          // [G]
    const int*   __restrict__ cids,          // [G, C]
    const int*   __restrict__ pids,          // [G, C]
    float*       __restrict__ out)           // [N_NODES, B] (pre-copied)
{
  __shared__ float tile[C * RSTRIDE];  // 64 gathered rows of 128 f32 (padded)
  __shared__ float wts[C];             // edge weights

  const int g    = blockIdx.x;
  const int tid  = threadIdx.x;
  const int lane = tid & 31;
  const int wv   = tid >> 5;           // wave id 0..7

  // ---- stage weights (params gather) ----
  if (tid < C) {
    wts[tid] = params[pids[(size_t)g * C + tid]];
  }

  // ---- async gather: 64 rows x 512B, one B128 per lane per row ----
  // wave wv stages rows [wv*8, wv*8+8)
  const uint32_t tile_base = (uint32_t)(uintptr_t)(&tile[0]);  // low 32 = LDS off
#pragma unroll
  for (int i = 0; i < 8; ++i) {
    const int c   = wv * 8 + i;
    const int cid = cids[(size_t)g * C + c];
    const float* src = element_mars + (size_t)cid * B + lane * 4;
    const uint32_t dst = tile_base + (uint32_t)(c * RSTRIDE + lane * 4) * 4u;
    async_row_load_b128(dst, src);
  }
  wait_async_zero();
  __syncthreads();

  // ---- per-column stable max over 64 children ----
  // wave wv owns columns [wv*16, wv*16+16); lane halves split the rows with a
  // 2-row offset so simultaneous LDS reads land 16 banks apart.
  const int half = lane >> 4;               // 0: lanes 0-15, 1: lanes 16-31
  const int col  = (wv << 4) + (lane & 15); // batch column this lane serves
  float m = -INFINITY;
#pragma unroll
  for (int i = 0; i < 16; ++i) {
    const int r = i * 4 + half * 2;
    m = fmaxf(m, tile[r * RSTRIDE + col]);
    m = fmaxf(m, tile[(r + 1) * RSTRIDE + col]);
  }
  m = fmaxf(m, __shfl_xor(m, 16, 32));      // merge the two row-halves

  // ---- weighted sum via WMMA f32 16x16x4, K-chained over 16 chunks ----
  // A(16x4): all 16 rows replicated with w[c0..c0+3]
  //   lanes 0-15 supply K0,K1 (a.x,a.y); lanes 16-31 supply K2,K3.
  // B(4x16): exp(ch - m) for the matching K rows at this lane's column.
  // D rows are all identical; row M=0 lives in acc[0] on lanes 0-15.
  v8f acc = {};
#pragma unroll
  for (int kk = 0; kk < 16; ++kk) {
    const int c0 = kk * 4 + half * 2;       // K rows this lane-half provides
    v2f a, b;
    a.x = wts[c0];
    a.y = wts[c0 + 1];
    b.x = __expf(tile[c0 * RSTRIDE + col] - m);
    b.y = __expf(tile[(c0 + 1) * RSTRIDE + col] - m);
    acc = __builtin_amdgcn_wmma_f32_16x16x4_f32(
        /*neg_a=*/false, a, /*neg_b=*/false, b,
        /*c_mod=*/(short)0, acc, /*reuse_a=*/false, /*reuse_b=*/false);
  }

  // ---- log(s) + m, scatter into the node marginal buffer ----
  if (lane < 16) {
    const float s = acc[0];                 // D[M=0, N=col]
    out[(size_t)nids[g] * B + col] = __logf(s) + m;
  }
}

extern "C" void kernel_launch(void* const* d_in, const int* in_sizes, int n_in,
                              void* d_out, int out_size, void* d_ws, size_t ws_size,
                              hipStream_t stream) {
  const float* node_mars    = (const float*)d_in[0];  // [N_NODES, B]
  const float* element_mars = (const float*)d_in[1];  // [MAX_ELS, B]
  const float* params       = (const float*)d_in[2];  // [N_PARAMS]
  const int*   nids         = (const int*)d_in[3];    // [G]
  const int*   cids         = (const int*)d_in[4];    // [G, C]
  const int*   pids         = (const int*)d_in[5];    // [G, C]
  float*       out          = (float*)d_out;

  const int G = in_sizes[3];

  // out = node_mars, then scatter the G updated rows (stream-ordered).
  hipMemcpyAsync(out, node_mars, (size_t)in_sizes[0] * sizeof(float),
                 hipMemcpyDeviceToDevice, stream);

  sum_layer_lse_kernel<<<G, 256, 0, stream>>>(element_mars, params, nids, cids,
                                              pids, out);
}